// GraphAutoencoder_31233002177120
// MI455X (gfx1250) — compile-verified
//
#include <hip/hip_runtime.h>
#include <math.h>

typedef __attribute__((ext_vector_type(2))) float v2f;
typedef __attribute__((ext_vector_type(8))) float v8f;

#define N_NODES 8192
#define IN_FEAT 128
#define OUT_FEAT 64
#define N_EDGES 262144

// ---------------- GCN encoder stages ----------------

__global__ void deg_init_kernel(float* deg, float* z) {
    int t = blockIdx.x * blockDim.x + threadIdx.x;
    if (t < N_NODES) deg[t] = 1.0f;           // self-loop contributes 1 to every node
    // zero z ([N_NODES, OUT_FEAT]) in the same pass (64x more elements)
    for (int i = t; i < N_NODES * OUT_FEAT; i += gridDim.x * blockDim.x) z[i] = 0.0f;
}

__global__ void deg_scatter_kernel(const long long* ei, float* deg) {
    int e = blockIdx.x * blockDim.x + threadIdx.x;
    if (e < N_EDGES) {
        int d = (int)ei[N_EDGES + e];         // edge_index[1][e]
        atomicAdd(&deg[d], 1.0f);
    }
}

__global__ void dinv_kernel(const float* deg, float* dinv) {
    int n = blockIdx.x * blockDim.x + threadIdx.x;
    if (n < N_NODES) dinv[n] = rsqrtf(deg[n]);
}

// h[n][f] = sum_k x[k][n] * W[k][f]   (x stored [IN_FEAT, N_NODES])
__global__ void feat_gemm_kernel(const float* __restrict__ x,
                                 const float* __restrict__ W,
                                 float* __restrict__ h) {
    int t = blockIdx.x * blockDim.x + threadIdx.x;   // N_NODES * OUT_FEAT threads
    int n = t >> 6;
    int f = t & 63;
    float acc = 0.0f;
#pragma unroll 8
    for (int k = 0; k < IN_FEAT; ++k)
        acc += x[k * N_NODES + n] * W[k * OUT_FEAT + f];
    h[t] = acc;
}

// z[dst] += h[src] * dinv[src]*dinv[dst], one thread per (edge, feature)
__global__ void edge_scatter_kernel(const long long* __restrict__ ei,
                                    const float* __restrict__ h,
                                    const float* __restrict__ dinv,
                                    float* __restrict__ z) {
    int t = blockIdx.x * blockDim.x + threadIdx.x;   // N_EDGES * 64 threads
    int e = t >> 6;
    int f = t & 63;
    int s = (int)ei[e];
    int d = (int)ei[N_EDGES + e];
    float norm = dinv[s] * dinv[d];
    atomicAdd(&z[d * OUT_FEAT + f], h[s * OUT_FEAT + f] * norm);
}

// self-loop term + bias
__global__ void self_bias_kernel(const float* __restrict__ h,
                                 const float* __restrict__ dinv,
                                 const float* __restrict__ b,
                                 float* __restrict__ z) {
    int t = blockIdx.x * blockDim.x + threadIdx.x;   // N_NODES * 64 threads
    int n = t >> 6;
    int f = t & 63;
    float di = dinv[n];
    z[t] += h[t] * di * di + b[f];
}

// ---------------- Decoder: sigmoid(z @ z^T) via V_WMMA_F32_16X16X4_F32 ----------------
// One wave -> one 32x32 output macro-tile (2x2 WMMA 16x16 tiles), K=64 in 16 steps of 4.
// Output (256 MB) is write-once streaming data and exceeds the 192 MB L2:
// store it non-temporally so the z operand set (2 MB) stays cache-resident.

__global__ __launch_bounds__(256)
void decoder_wmma_kernel(const float* __restrict__ z, float* __restrict__ out) {
    const int lane  = threadIdx.x & 31;
    const int wave  = (blockIdx.x << 3) + (threadIdx.x >> 5);  // 8 waves / block
    const int tileRow = wave >> 8;          // 256 macro-tile columns (8192/32)
    const int tileCol = wave & 255;
    const int rowBase = tileRow << 5;
    const int colBase = tileCol << 5;

    const int lidx  = lane & 15;            // M (for A) / N (for B) within tile
    const int khalf = (lane >> 4) << 1;     // lanes 0-15 hold K=k0,k0+1; lanes 16-31 hold K=k0+2,k0+3

    v8f acc00 = {}, acc01 = {}, acc10 = {}, acc11 = {};

    const float* aRow0 = z + (rowBase +      lidx) * OUT_FEAT + khalf;
    const float* aRow1 = z + (rowBase + 16 + lidx) * OUT_FEAT + khalf;
    const float* bRow0 = z + (colBase +      lidx) * OUT_FEAT + khalf;
    const float* bRow1 = z + (colBase + 16 + lidx) * OUT_FEAT + khalf;

#pragma unroll
    for (int k0 = 0; k0 < OUT_FEAT; k0 += 4) {
        v2f a0 = *(const v2f*)(aRow0 + k0);
        v2f a1 = *(const v2f*)(aRow1 + k0);
        v2f b0 = *(const v2f*)(bRow0 + k0);
        v2f b1 = *(const v2f*)(bRow1 + k0);
        acc00 = __builtin_amdgcn_wmma_f32_16x16x4_f32(false, a0, false, b0, (short)0, acc00, false, false);
        acc01 = __builtin_amdgcn_wmma_f32_16x16x4_f32(false, a0, false, b1, (short)0, acc01, false, false);
        acc10 = __builtin_amdgcn_wmma_f32_16x16x4_f32(false, a1, false, b0, (short)0, acc10, false, false);
        acc11 = __builtin_amdgcn_wmma_f32_16x16x4_f32(false, a1, false, b1, (short)0, acc11, false, false);
    }

    // C/D layout: VGPR v -> M = v + 8*(lane>=16), N = lane&15
    const int mOff = (lane >> 4) << 3;      // 0 or 8
    v8f accs[4] = {acc00, acc01, acc10, acc11};
#pragma unroll
    for (int rt = 0; rt < 2; ++rt) {
#pragma unroll
        for (int ct = 0; ct < 2; ++ct) {
            v8f c = accs[rt * 2 + ct];
            int row0 = rowBase + rt * 16 + mOff;
            int col  = colBase + ct * 16 + lidx;
#pragma unroll
            for (int v = 0; v < 8; ++v) {
                float s = 1.0f / (1.0f + __expf(-c[v]));
                __builtin_nontemporal_store(s, &out[(size_t)(row0 + v) * N_NODES + col]);
            }
        }
    }
}

// ---------------- Launch ----------------

extern "C" void kernel_launch(void* const* d_in, const int* in_sizes, int n_in,
                              void* d_out, int out_size, void* d_ws, size_t ws_size,
                              hipStream_t stream) {
    const float*     x  = (const float*)d_in[0];      // [128, 8192]
    const long long* ei = (const long long*)d_in[1];  // [2, 262144] int64
    const float*     W  = (const float*)d_in[2];      // [128, 64]
    const float*     b  = (const float*)d_in[3];      // [64]
    float* out = (float*)d_out;                       // [8192, 8192]

    float* deg  = (float*)d_ws;                       // 8192
    float* dinv = deg + N_NODES;                      // 8192
    float* h    = dinv + N_NODES;                     // 8192*64
    float* z    = h + N_NODES * OUT_FEAT;             // 8192*64

    // 1. degree (with self-loops) + zero z
    deg_init_kernel<<<N_NODES / 256, 256, 0, stream>>>(deg, z);
    deg_scatter_kernel<<<N_EDGES / 256, 256, 0, stream>>>(ei, deg);
    dinv_kernel<<<N_NODES / 256, 256, 0, stream>>>(deg, dinv);

    // 2. h = x^T @ W
    feat_gemm_kernel<<<(N_NODES * OUT_FEAT) / 256, 256, 0, stream>>>(x, W, h);

    // 3. z = scatter(h * norm) + self-loop + bias
    edge_scatter_kernel<<<(N_EDGES * OUT_FEAT) / 256, 256, 0, stream>>>(ei, h, dinv, z);
    self_bias_kernel<<<(N_NODES * OUT_FEAT) / 256, 256, 0, stream>>>(h, dinv, b, z);

    // 4. out = sigmoid(z @ z^T): (8192/32)^2 = 65536 waves, 8 waves/block
    decoder_wmma_kernel<<<65536 / 8, 256, 0, stream>>>(z, out);
}